// PatientMeanEncoder_50328426775181
// MI455X (gfx1250) — compile-verified
//
#include <hip/hip_runtime.h>
#include <hip/hip_bf16.h>

typedef __attribute__((ext_vector_type(2))) float v2f;
typedef __attribute__((ext_vector_type(8))) float v8f;

#define NB     64
#define LSEQ   2048
#define CIN    256
#define DEMF   10
#define COUT   276   // 256 + 20
#define CPAD   288   // COUT padded to multiple of 32
#define NCHUNK 16
#define CHUNK  128   // LSEQ / NCHUNK

// ---------------------------------------------------------------------------
// Kernel 0: dem MLP on the WMMA path.
//   H  = relu(dem @ W1 + b1)   M=64, K=10 (pad 12), N=40 (pad 48)
//   E  = relu(H   @ W2 + b2)   M=64, K=40 (pad 48), N=20 (pad 32)
// One workgroup, 4 waves; wave w owns output rows [16w,16w+16).
// A-frag (16x4 f32): lane m=lane%16; vgpr0/1 = K {0,1} (lanes 0-15) / {2,3}.
// B-frag (4x16 f32): mirrored K striping, N = lane%16.
// C/D (16x16 f32): vgpr r -> row r (lanes 0-15) / r+8 (lanes 16-31).
// All-lane execution around every WMMA (zero-fill via selects, no branches).
// ---------------------------------------------------------------------------
__global__ __launch_bounds__(128) void mlp_wmma_kernel(
    const float* __restrict__ dem, const float* __restrict__ W1,
    const float* __restrict__ b1, const float* __restrict__ W2,
    const float* __restrict__ b2, float* __restrict__ demE) {
  __shared__ float H[64][48];
  const int lane = threadIdx.x & 31;
  const int wave = threadIdx.x >> 5;
  const int half = lane >> 4;          // 0 -> K{0,1}, 1 -> K{2,3}
  const int mrow = wave * 16 + (lane & 15);
  const int ncol = lane & 15;

  // ----- layer 1 -----
  for (int nt = 0; nt < 3; ++nt) {
    v8f acc = {};
    const int col = nt * 16 + ncol;
#pragma unroll
    for (int ks = 0; ks < 3; ++ks) {
      const int k0 = ks * 4 + half * 2;
      v2f a, b;
      a.x = (k0     < DEMF) ? dem[mrow * DEMF + k0]     : 0.f;
      a.y = (k0 + 1 < DEMF) ? dem[mrow * DEMF + k0 + 1] : 0.f;
      b.x = (k0     < DEMF && col < 40) ? W1[k0 * 40 + col]       : 0.f;
      b.y = (k0 + 1 < DEMF && col < 40) ? W1[(k0 + 1) * 40 + col] : 0.f;
      acc = __builtin_amdgcn_wmma_f32_16x16x4_f32(false, a, false, b,
                                                  (short)0, acc, false, false);
    }
    const float bias = (col < 40) ? b1[col] : 0.f;
#pragma unroll
    for (int r = 0; r < 8; ++r) {
      const int row = wave * 16 + half * 8 + r;
      H[row][col] = fmaxf(acc[r] + bias, 0.f);   // padded cols -> exactly 0
    }
  }
  __syncthreads();

  // ----- layer 2 -----
  for (int nt = 0; nt < 2; ++nt) {
    v8f acc = {};
    const int col = nt * 16 + ncol;
#pragma unroll
    for (int ks = 0; ks < 12; ++ks) {
      const int k0 = ks * 4 + half * 2;
      v2f a, b;
      a.x = H[mrow][k0];
      a.y = H[mrow][k0 + 1];
      b.x = (k0     < 40 && col < 20) ? W2[k0 * 20 + col]       : 0.f;
      b.y = (k0 + 1 < 40 && col < 20) ? W2[(k0 + 1) * 20 + col] : 0.f;
      acc = __builtin_amdgcn_wmma_f32_16x16x4_f32(false, a, false, b,
                                                  (short)0, acc, false, false);
    }
    const float bias = (col < 20) ? b2[col] : 0.f;
#pragma unroll
    for (int r = 0; r < 8; ++r) {
      const int row = wave * 16 + half * 8 + r;
      demE[row * 32 + col] = fmaxf(acc[r] + bias, 0.f);  // stride-32 padded
    }
  }
}

// ---------------------------------------------------------------------------
// Phase 1: per-(n, chunk, c) local sum and nonzero count over CHUNK steps.
// Lane index == channel -> coalesced B32 streams from timesteps.
// ---------------------------------------------------------------------------
__global__ __launch_bounds__(CPAD) void phase1_kernel(
    const float* __restrict__ ts, const float* __restrict__ demE,
    float* __restrict__ psum, float* __restrict__ pcnt) {
  const int n  = blockIdx.x / NCHUNK;
  const int ch = blockIdx.x % NCHUNK;
  const int c  = threadIdx.x;
  float s = 0.f, k = 0.f;
  if (c < COUT) {
    if (c < CIN) {
      const float* p = ts + ((size_t)n * LSEQ + (size_t)ch * CHUNK) * CIN + c;
#pragma unroll 4
      for (int i = 0; i < CHUNK; ++i) {
        const float v = p[(size_t)i * CIN];
        s += v;
        k += (v != 0.f) ? 1.f : 0.f;
      }
    } else {
      const float v = demE[n * 32 + (c - CIN)];   // constant over L
      s = v * (float)CHUNK;
      k = (v != 0.f) ? (float)CHUNK : 0.f;
    }
  }
  const size_t o = ((size_t)n * NCHUNK + ch) * CPAD + c;
  psum[o] = s;
  pcnt[o] = k;
}

// ---------------------------------------------------------------------------
// Phase 2: exclusive prefix over the 16 chunk partials, per (n, c), in place.
// ---------------------------------------------------------------------------
__global__ __launch_bounds__(CPAD) void phase2_kernel(
    float* __restrict__ psum, float* __restrict__ pcnt) {
  const int n = blockIdx.x;
  const int c = threadIdx.x;
  float rs = 0.f, rk = 0.f;
  for (int ch = 0; ch < NCHUNK; ++ch) {
    const size_t o = ((size_t)n * NCHUNK + ch) * CPAD + c;
    const float s = psum[o];
    const float k = pcnt[o];
    psum[o] = rs;
    pcnt[o] = rk;
    rs += s;
    rk += k;
  }
}

// ---------------------------------------------------------------------------
// Phase 3: re-scan each chunk with its exclusive prefix; timesteps re-read
// hits the 192 MB L2 (the whole 134 MB tensor is resident). Coalesced stores.
// ---------------------------------------------------------------------------
__global__ __launch_bounds__(CPAD) void phase3_kernel(
    const float* __restrict__ ts, const float* __restrict__ demE,
    const float* __restrict__ psum, const float* __restrict__ pcnt,
    float* __restrict__ out) {
  const int n  = blockIdx.x / NCHUNK;
  const int ch = blockIdx.x % NCHUNK;
  const int c  = threadIdx.x;
  if (c >= COUT) return;

  const size_t po = ((size_t)n * NCHUNK + ch) * CPAD + c;
  float s   = psum[po];
  float cnt = pcnt[po];
  float* op = out + ((size_t)n * LSEQ + (size_t)ch * CHUNK) * COUT + c;

  if (c < CIN) {
    const float* p = ts + ((size_t)n * LSEQ + (size_t)ch * CHUNK) * CIN + c;
#pragma unroll 4
    for (int i = 0; i < CHUNK; ++i) {
      const float v = p[(size_t)i * CIN];
      s   += v;
      cnt += (v != 0.f) ? 1.f : 0.f;
      op[(size_t)i * COUT] = fmaxf(s / fmaxf(cnt, 1.f), 0.f);
    }
  } else {
    const float v = demE[n * 32 + (c - CIN)];
    const float dc = (v != 0.f) ? 1.f : 0.f;
#pragma unroll 4
    for (int i = 0; i < CHUNK; ++i) {
      s   += v;
      cnt += dc;
      op[(size_t)i * COUT] = fmaxf(s / fmaxf(cnt, 1.f), 0.f);
    }
  }
}

// ---------------------------------------------------------------------------
extern "C" void kernel_launch(void* const* d_in, const int* in_sizes, int n_in,
                              void* d_out, int out_size, void* d_ws, size_t ws_size,
                              hipStream_t stream) {
  const float* ts  = (const float*)d_in[0];  // (64, 2048, 256)
  const float* dem = (const float*)d_in[1];  // (64, 10)
  const float* W1  = (const float*)d_in[2];  // (10, 40)
  const float* b1  = (const float*)d_in[3];  // (40,)
  const float* W2  = (const float*)d_in[4];  // (40, 20)
  const float* b2  = (const float*)d_in[5];  // (20,)
  float* out = (float*)d_out;                // (64, 2048, 276)

  float* ws   = (float*)d_ws;
  float* demE = ws;                                   // 64*32 floats (padded)
  float* psum = ws + 2048;                            // 64*16*288
  float* pcnt = psum + (size_t)NB * NCHUNK * CPAD;    // 64*16*288

  mlp_wmma_kernel<<<1, 128, 0, stream>>>(dem, W1, b1, W2, b2, demE);
  phase1_kernel<<<NB * NCHUNK, CPAD, 0, stream>>>(ts, demE, psum, pcnt);
  phase2_kernel<<<NB, CPAD, 0, stream>>>(psum, pcnt);
  phase3_kernel<<<NB * NCHUNK, CPAD, 0, stream>>>(ts, demE, psum, pcnt, out);
}